// LiteEIISRS_87866440942259
// MI455X (gfx1250) — compile-verified
//
#include <hip/hip_runtime.h>

typedef __attribute__((ext_vector_type(2))) float v2f;
typedef __attribute__((ext_vector_type(8))) float v8f;

#define N_USERS   100000
#define N_ITEMS   50000
#define N_TOTAL   150000
#define EMB_D     64
#define ALPHA     0.1f
#define ROW_TILES (N_TOTAL / 16)      // 9375
#define COL_TILES (EMB_D / 16)        // 4
#define TILES_TOTAL (3 * ROW_TILES * COL_TILES)
#define LAYER_ELEMS ((size_t)N_TOTAL * EMB_D)

// ---------------------------------------------------------------------------
// Dense init: out[layer] = scale * emb0 (scale = 1.0 layer 0, ALPHA layers 1,2)
// per 16x16 tile with V_WMMA_F32_16X16X4_F32:
//   D = sum_{j=0..3} A_j(16x4) x B_j(4x16), A_j = scale * identity band.
// Exact f32 math (each D element is exactly one multiply).
// One wave per tile; tile id is wave-uniform so EXEC stays all-ones.
// ---------------------------------------------------------------------------
__global__ void lgcn_init_wmma(const float* __restrict__ u0,
                               const float* __restrict__ i0,
                               float* __restrict__ out)
{
    const int lane = threadIdx.x & 31;
    const int wave = threadIdx.x >> 5;
    const int tile = blockIdx.x * (blockDim.x >> 5) + wave;
    if (tile >= TILES_TOTAL) return;          // wave-uniform exit

    const int colTile  = tile & (COL_TILES - 1);
    const int rowTileG = tile >> 2;
    const int layer    = rowTileG / ROW_TILES;
    const int rowTile  = rowTileG % ROW_TILES;
    const int row0     = rowTile * 16;
    const int col0     = colTile * 16;
    const float scale  = (layer == 0) ? 1.0f : ALPHA;

    const float* src;
    int srcRow0;
    if (row0 < N_USERS) { src = u0; srcRow0 = row0; }
    else                { src = i0; srcRow0 = row0 - N_USERS; }

    const int n    = lane & 15;   // column within tile (B/D layout)
    const int m    = lane & 15;   // row index for A layout
    const int half = lane >> 4;   // lanes 16-31 carry K=2,3 slots

    v8f acc = {};
    #pragma unroll
    for (int j = 0; j < 4; ++j) {
        const int k0 = 4 * j + (half ? 2 : 0);
        const int k1 = k0 + 1;
        v2f b;
        b.x = src[(size_t)(srcRow0 + k0) * EMB_D + col0 + n];
        b.y = src[(size_t)(srcRow0 + k1) * EMB_D + col0 + n];
        v2f a;
        a.x = (m == k0) ? scale : 0.0f;
        a.y = (m == k1) ? scale : 0.0f;
        acc = __builtin_amdgcn_wmma_f32_16x16x4_f32(
                  false, a, false, b, (short)0, acc, false, false);
    }

    float* dst = out + (size_t)layer * LAYER_ELEMS
                     + (size_t)row0 * EMB_D + col0;
    #pragma unroll
    for (int v = 0; v < 8; ++v) {
        const int M = v + half * 8;
        dst[(size_t)M * EMB_D + n] = acc[v];
    }
}

// ---------------------------------------------------------------------------
// Counting-sort (COO -> CSR) helpers
// ---------------------------------------------------------------------------
__global__ void lgcn_zero_u32(unsigned* __restrict__ p, int n)
{
    int i = blockIdx.x * blockDim.x + threadIdx.x;
    if (i < n) p[i] = 0u;
}

__global__ void lgcn_hist_rows(const int* __restrict__ rows,
                               unsigned* __restrict__ cnt, int nnz)
{
    int e = blockIdx.x * blockDim.x + threadIdx.x;
    if (e < nnz) atomicAdd(&cnt[rows[e]], 1u);
}

// Single-workgroup chunked exclusive scan: start[i] = sum_{j<i} cnt[j],
// start[n] = total.  1024 threads, Hillis-Steele per 1024-chunk with carry.
__global__ void lgcn_exscan(const unsigned* __restrict__ cnt,
                            unsigned* __restrict__ start, int n)
{
    __shared__ unsigned buf[1024];
    __shared__ unsigned carry;
    const int tid = threadIdx.x;
    if (tid == 0) carry = 0u;
    __syncthreads();

    for (int base = 0; base < n; base += 1024) {
        const int idx = base + tid;
        const unsigned v = (idx < n) ? cnt[idx] : 0u;
        buf[tid] = v;
        __syncthreads();
        for (int off = 1; off < 1024; off <<= 1) {
            unsigned t = (tid >= off) ? buf[tid - off] : 0u;
            __syncthreads();
            buf[tid] += t;
            __syncthreads();
        }
        const unsigned c = carry;
        if (idx < n) start[idx] = c + buf[tid] - v;   // exclusive
        __syncthreads();
        if (tid == 0) carry = c + buf[1023];
        __syncthreads();
    }
    if (tid == 0) start[n] = carry;                    // total == nnz
}

__global__ void lgcn_copy_u32(const unsigned* __restrict__ src,
                              unsigned* __restrict__ dst, int n)
{
    int i = blockIdx.x * blockDim.x + threadIdx.x;
    if (i < n) dst[i] = src[i];
}

__global__ void lgcn_scatter(const int* __restrict__ rows,
                             const int* __restrict__ cols,
                             const float* __restrict__ vals,
                             unsigned* __restrict__ cursor,
                             int*   __restrict__ scols,
                             float* __restrict__ svals, int nnz)
{
    int e = blockIdx.x * blockDim.x + threadIdx.x;
    if (e >= nnz) return;
    const unsigned d = atomicAdd(&cursor[rows[e]], 1u);
    scols[d] = cols[e];
    svals[d] = vals[e];
}

// ---------------------------------------------------------------------------
// CSR gather SpMM: y[r,:] += sum_k svals[k] * x[scols[k],:]
// 16 lanes own one row (float4 accumulators in registers, 256B coalesced
// gathers per half-wave).  No atomics; each row written exactly once, folded
// onto the ALPHA residual written by lgcn_init_wmma.
// ---------------------------------------------------------------------------
__global__ void lgcn_spmm_csr(const unsigned* __restrict__ start,
                              const int*      __restrict__ scols,
                              const float*    __restrict__ svals,
                              const float*    __restrict__ x,
                              float*          __restrict__ y,
                              int nrows)
{
    const int t = blockIdx.x * blockDim.x + threadIdx.x;
    const int r = t >> 4;
    if (r >= nrows) return;
    const int sub = t & 15;

    const unsigned kb = start[r];
    const unsigned ke = start[r + 1];

    float4 acc = make_float4(0.f, 0.f, 0.f, 0.f);
    for (unsigned k = kb; k < ke; ++k) {
        const int   c = scols[k];
        const float v = svals[k];
        const float4 xv = ((const float4*)(x + (size_t)c * EMB_D))[sub];
        acc.x += v * xv.x;
        acc.y += v * xv.y;
        acc.z += v * xv.z;
        acc.w += v * xv.w;
    }

    float4* yp = (float4*)(y + (size_t)r * EMB_D) + sub;
    float4 cur = *yp;                 // ALPHA * emb0 residual
    cur.x += acc.x; cur.y += acc.y; cur.z += acc.z; cur.w += acc.w;
    *yp = cur;
}

// ---------------------------------------------------------------------------
// Fallback (small workspace): COO scatter with fp32 atomics.
// ---------------------------------------------------------------------------
__global__ void lgcn_spmm_atomic(const int*   __restrict__ rows,
                                 const int*   __restrict__ cols,
                                 const float* __restrict__ vals,
                                 const float* __restrict__ x,
                                 float*       __restrict__ y,
                                 int nnz)
{
    const int t = blockIdx.x * blockDim.x + threadIdx.x;
    const int e = t >> 4;
    if (e >= nnz) return;
    const int sub = t & 15;

    const int   r = rows[e];
    const int   c = cols[e];
    const float v = vals[e];

    const float4 xv = ((const float4*)(x + (size_t)c * EMB_D))[sub];
    float* yp = y + (size_t)r * EMB_D + sub * 4;
    atomicAdd(yp + 0, v * xv.x);
    atomicAdd(yp + 1, v * xv.y);
    atomicAdd(yp + 2, v * xv.z);
    atomicAdd(yp + 3, v * xv.w);
}

static inline size_t align16(size_t x) { return (x + 15) & ~(size_t)15; }

extern "C" void kernel_launch(void* const* d_in, const int* in_sizes, int n_in,
                              void* d_out, int out_size, void* d_ws, size_t ws_size,
                              hipStream_t stream) {
    const float* u0      = (const float*)d_in[0];
    const float* i0      = (const float*)d_in[1];
    const float* ui_vals = (const float*)d_in[2];
    const float* iu_vals = (const float*)d_in[3];
    const int*   ui_rows = (const int*)d_in[4];
    const int*   ui_cols = (const int*)d_in[5];
    const int*   iu_rows = (const int*)d_in[6];
    const int*   iu_cols = (const int*)d_in[7];
    float* out = (float*)d_out;
    (void)n_in; (void)out_size;

    const int nnz_ui = in_sizes[2];
    const int nnz_iu = in_sizes[3];

    // 1) Dense init via WMMA: layer0 = emb0, layers 1,2 = ALPHA*emb0
    {
        const int wavesPerBlock = 8;                   // 256 threads, wave32
        const int blocks = (TILES_TOTAL + wavesPerBlock - 1) / wavesPerBlock;
        lgcn_init_wmma<<<blocks, 256, 0, stream>>>(u0, i0, out);
    }

    float* u1 = out + LAYER_ELEMS;
    float* i1 = out + LAYER_ELEMS + (size_t)N_USERS * EMB_D;
    float* u2 = out + 2 * LAYER_ELEMS;
    float* i2 = out + 2 * LAYER_ELEMS + (size_t)N_USERS * EMB_D;

    // Workspace layout for the CSR path
    size_t off = 0;
    const size_t o_ui_start  = off; off += align16(sizeof(unsigned) * (N_USERS + 1));
    const size_t o_ui_cursor = off; off += align16(sizeof(unsigned) * N_USERS);
    const size_t o_iu_start  = off; off += align16(sizeof(unsigned) * (N_ITEMS + 1));
    const size_t o_iu_cursor = off; off += align16(sizeof(unsigned) * N_ITEMS);
    const size_t o_ui_scols  = off; off += align16(sizeof(int)   * (size_t)nnz_ui);
    const size_t o_ui_svals  = off; off += align16(sizeof(float) * (size_t)nnz_ui);
    const size_t o_iu_scols  = off; off += align16(sizeof(int)   * (size_t)nnz_iu);
    const size_t o_iu_svals  = off; off += align16(sizeof(float) * (size_t)nnz_iu);
    const size_t ws_needed = off;

    if (ws_size >= ws_needed) {
        char* ws = (char*)d_ws;
        unsigned* ui_start  = (unsigned*)(ws + o_ui_start);
        unsigned* ui_cursor = (unsigned*)(ws + o_ui_cursor);
        unsigned* iu_start  = (unsigned*)(ws + o_iu_start);
        unsigned* iu_cursor = (unsigned*)(ws + o_iu_cursor);
        int*      ui_scols  = (int*)     (ws + o_ui_scols);
        float*    ui_svals  = (float*)   (ws + o_ui_svals);
        int*      iu_scols  = (int*)     (ws + o_iu_scols);
        float*    iu_svals  = (float*)   (ws + o_iu_svals);

        const int gE_ui = (nnz_ui + 255) / 256;
        const int gE_iu = (nnz_iu + 255) / 256;
        const int gU = (N_USERS + 255) / 256;
        const int gI = (N_ITEMS + 255) / 256;

        // --- build CSR for A_ui (rows in [0, N_USERS)) ---
        lgcn_zero_u32 <<<gU, 256, 0, stream>>>(ui_cursor, N_USERS);
        lgcn_hist_rows<<<gE_ui, 256, 0, stream>>>(ui_rows, ui_cursor, nnz_ui);
        lgcn_exscan   <<<1, 1024, 0, stream>>>(ui_cursor, ui_start, N_USERS);
        lgcn_copy_u32 <<<gU, 256, 0, stream>>>(ui_start, ui_cursor, N_USERS);
        lgcn_scatter  <<<gE_ui, 256, 0, stream>>>(ui_rows, ui_cols, ui_vals,
                                                  ui_cursor, ui_scols, ui_svals, nnz_ui);
        // --- build CSR for A_iu (rows in [0, N_ITEMS)) ---
        lgcn_zero_u32 <<<gI, 256, 0, stream>>>(iu_cursor, N_ITEMS);
        lgcn_hist_rows<<<gE_iu, 256, 0, stream>>>(iu_rows, iu_cursor, nnz_iu);
        lgcn_exscan   <<<1, 1024, 0, stream>>>(iu_cursor, iu_start, N_ITEMS);
        lgcn_copy_u32 <<<gI, 256, 0, stream>>>(iu_start, iu_cursor, N_ITEMS);
        lgcn_scatter  <<<gE_iu, 256, 0, stream>>>(iu_rows, iu_cols, iu_vals,
                                                  iu_cursor, iu_scols, iu_svals, nnz_iu);

        // --- atomic-free gather SpMMs (stream order = layer dependency) ---
        const int gRu = (N_USERS * 16 + 255) / 256;
        const int gRi = (N_ITEMS * 16 + 255) / 256;
        lgcn_spmm_csr<<<gRu, 256, 0, stream>>>(ui_start, ui_scols, ui_svals, i0, u1, N_USERS);
        lgcn_spmm_csr<<<gRi, 256, 0, stream>>>(iu_start, iu_scols, iu_svals, u0, i1, N_ITEMS);
        lgcn_spmm_csr<<<gRu, 256, 0, stream>>>(ui_start, ui_scols, ui_svals, i1, u2, N_USERS);
        lgcn_spmm_csr<<<gRi, 256, 0, stream>>>(iu_start, iu_scols, iu_svals, u1, i2, N_ITEMS);
    } else {
        // Workspace too small: COO scatter with fp32 atomics.
        const int g_ui = (nnz_ui * 16 + 255) / 256;
        const int g_iu = (nnz_iu * 16 + 255) / 256;
        lgcn_spmm_atomic<<<g_ui, 256, 0, stream>>>(ui_rows, ui_cols, ui_vals, i0, u1, nnz_ui);
        lgcn_spmm_atomic<<<g_iu, 256, 0, stream>>>(iu_rows, iu_cols, iu_vals, u0, i1, nnz_iu);
        lgcn_spmm_atomic<<<g_ui, 256, 0, stream>>>(ui_rows, ui_cols, ui_vals, i1, u2, nnz_ui);
        lgcn_spmm_atomic<<<g_iu, 256, 0, stream>>>(iu_rows, iu_cols, iu_vals, u1, i2, nnz_iu);
    }
}